// SupervisedContrastiveLoss_5600637354498
// MI455X (gfx1250) — compile-verified
//
#include <hip/hip_runtime.h>
#include <hip/hip_bf16.h>

// Supervised contrastive loss, N=8192, D=128, fp32, T=0.07.
// sim tiles via V_WMMA_F32_16X16X4_F32; each wave owns TWO i-tiles (32 rows)
// so every B fragment load feeds two independent WMMAs (2x data reuse).

typedef __attribute__((ext_vector_type(2))) float v2f;
typedef __attribute__((ext_vector_type(8))) float v8f;

#define N_EMB    8192
#define DIM      128
#define NTILES   (N_EMB / 16)     // 512 j-tiles
#define NWAVES   4
#define ROWS_BLK 32               // rows per block (2 i-tiles, shared by all waves)
#define NBLOCKS  (N_EMB / ROWS_BLK)   // 256
#define INV_T    (1.0f / 0.07f)   // constant softmax shift C = 1/T (== sim_ii)

__global__ __launch_bounds__(128) void scl_tile_kernel(
    const float* __restrict__ emb,
    const int*   __restrict__ labels,
    float*       __restrict__ blocksum)
{
    const int tid  = threadIdx.x;
    const int wave = tid >> 5;
    const int lane = tid & 31;
    const int m    = lane & 15;   // row/col within tile
    const int h    = lane >> 4;   // half-wave: selects K pair (2h, 2h+1)
    const int i0   = blockIdx.x * ROWS_BLK;

    // ---- Preload A fragments for both i-tiles, full K=128 ----
    // WMMA 16x4 f32 A layout: lanes 0-15 hold K=0,1 ; lanes 16-31 hold K=2,3
    v2f a0[32], a1[32];
    const float* arow0 = emb + (size_t)(i0 + m) * DIM + 2 * h;
    const float* arow1 = emb + (size_t)(i0 + 16 + m) * DIM + 2 * h;
#pragma unroll
    for (int t = 0; t < 32; ++t) {
        a0[t] = *(const v2f*)(arow0 + 4 * t);
        a1[t] = *(const v2f*)(arow1 + 4 * t);
    }

    // Labels of the rows this lane's accumulator slots correspond to
    const int ib0 = i0 + 8 * h;          // tile0: slot r -> row ib0 + r
    const int ib1 = i0 + 16 + 8 * h;     // tile1: slot r -> row ib1 + r
    int li0[8], li1[8];
#pragma unroll
    for (int r = 0; r < 8; ++r) {
        li0[r] = labels[ib0 + r];
        li1[r] = labels[ib1 + r];
    }

    // Per-slot running stats (shift-invariant softmax, constant shift C)
    float d0[8], p0[8], c0[8], d1[8], p1[8], c1[8];
#pragma unroll
    for (int r = 0; r < 8; ++r) {
        d0[r] = 0.f; p0[r] = 0.f; c0[r] = 0.f;
        d1[r] = 0.f; p1[r] = 0.f; c1[r] = 0.f;
    }

    // ---- Sweep j-tiles (this wave handles every NWAVES-th tile) ----
    for (int jt = wave; jt < NTILES; jt += NWAVES) {
        const int j0   = jt * 16;
        const int jcol = j0 + m;                  // this lane's column index
        const int lj   = labels[jcol];
        const float* brow = emb + (size_t)jcol * DIM + 2 * h;

        v8f acc0 = {0.f, 0.f, 0.f, 0.f, 0.f, 0.f, 0.f, 0.f};
        v8f acc1 = {0.f, 0.f, 0.f, 0.f, 0.f, 0.f, 0.f, 0.f};
#pragma unroll
        for (int t = 0; t < 32; ++t) {
            v2f b = *(const v2f*)(brow + 4 * t);
            acc0 = __builtin_amdgcn_wmma_f32_16x16x4_f32(
                false, a0[t], false, b, (short)0, acc0, false, false);
            acc1 = __builtin_amdgcn_wmma_f32_16x16x4_f32(
                false, a1[t], false, b, (short)0, acc1, false, false);
        }

#pragma unroll
        for (int r = 0; r < 8; ++r) {
            // tile 0
            {
                const float s    = acc0[r] * INV_T;
                const bool  diag = (ib0 + r == jcol);
                float e = __expf(s - INV_T);
                if (diag) e = 0.f;
                d0[r] += e;
                if (!diag && (li0[r] == lj)) { p0[r] += s; c0[r] += 1.f; }
            }
            // tile 1
            {
                const float s    = acc1[r] * INV_T;
                const bool  diag = (ib1 + r == jcol);
                float e = __expf(s - INV_T);
                if (diag) e = 0.f;
                d1[r] += e;
                if (!diag && (li1[r] == lj)) { p1[r] += s; c1[r] += 1.f; }
            }
        }
    }

    // ---- Reduce across the 16 lanes of each half-wave (columns) ----
#pragma unroll
    for (int r = 0; r < 8; ++r) {
        float a_, b_, cc_;
        a_ = d0[r]; b_ = p0[r]; cc_ = c0[r];
#pragma unroll
        for (int mask = 1; mask < 16; mask <<= 1) {
            a_  += __shfl_xor(a_,  mask, 32);
            b_  += __shfl_xor(b_,  mask, 32);
            cc_ += __shfl_xor(cc_, mask, 32);
        }
        d0[r] = a_; p0[r] = b_; c0[r] = cc_;

        a_ = d1[r]; b_ = p1[r]; cc_ = c1[r];
#pragma unroll
        for (int mask = 1; mask < 16; mask <<= 1) {
            a_  += __shfl_xor(a_,  mask, 32);
            b_  += __shfl_xor(b_,  mask, 32);
            cc_ += __shfl_xor(cc_, mask, 32);
        }
        d1[r] = a_; p1[r] = b_; c1[r] = cc_;
    }

    // ---- Combine the 4 waves' partials in LDS ----
    __shared__ float red[NWAVES][ROWS_BLK][4];
    if (m == 0) {               // lanes 0 and 16 hold the reduced values
#pragma unroll
        for (int r = 0; r < 8; ++r) {
            red[wave][r + 8 * h][0]      = d0[r];
            red[wave][r + 8 * h][1]      = p0[r];
            red[wave][r + 8 * h][2]      = c0[r];
            red[wave][16 + r + 8 * h][0] = d1[r];
            red[wave][16 + r + 8 * h][1] = p1[r];
            red[wave][16 + r + 8 * h][2] = c1[r];
        }
    }
    __syncthreads();

    __shared__ float rterm[ROWS_BLK];
    if (tid < ROWS_BLK) {
        float d_ = 0.f, p_ = 0.f, c_ = 0.f;
#pragma unroll
        for (int w = 0; w < NWAVES; ++w) {
            d_ += red[w][tid][0];
            p_ += red[w][tid][1];
            c_ += red[w][tid][2];
        }
        const float logden = __logf(d_ + 1e-8f);
        rterm[tid] = (p_ - c_ * (INV_T + logden)) / fmaxf(c_, 1.0f);
    }
    __syncthreads();

    if (tid == 0) {
        float s = 0.f;
#pragma unroll
        for (int k = 0; k < ROWS_BLK; ++k) s += rterm[k];
        blocksum[blockIdx.x] = s;
    }
}

__global__ __launch_bounds__(NBLOCKS) void scl_final_kernel(
    const float* __restrict__ blocksum, float* __restrict__ out)
{
    __shared__ float sh[NBLOCKS];
    const int t = threadIdx.x;
    sh[t] = blocksum[t];
    __syncthreads();
    for (int off = NBLOCKS / 2; off > 0; off >>= 1) {
        if (t < off) sh[t] += sh[t + off];
        __syncthreads();
    }
    if (t == 0) out[0] = -sh[0] / (float)N_EMB;
}

extern "C" void kernel_launch(void* const* d_in, const int* in_sizes, int n_in,
                              void* d_out, int out_size, void* d_ws, size_t ws_size,
                              hipStream_t stream) {
    const float* emb    = (const float*)d_in[0];  // [8192,128] fp32, L2-normalized
    const int*   labels = (const int*)d_in[1];    // [8192] int32
    float*       out    = (float*)d_out;          // scalar loss
    float*       bsum   = (float*)d_ws;           // NBLOCKS per-block partials

    scl_tile_kernel<<<NBLOCKS, 128, 0, stream>>>(emb, labels, bsum);
    scl_final_kernel<<<1, NBLOCKS, 0, stream>>>(bsum, out);
}